// SAGE_26585847562498
// MI455X (gfx1250) — compile-verified
//
#include <hip/hip_runtime.h>

#define NN 50000       // nodes
#define NE 640000      // edges
#define FIN 20         // input features
#define HID 128        // hidden
#define NG 100         // graphs
#define NOUT 2         // classes

typedef __attribute__((ext_vector_type(2))) float v2f;
typedef __attribute__((ext_vector_type(8))) float v8f;

// ---------------- workspace layout (float offsets, all 16-float aligned) ---
static const size_t OFF_AGG1 = 0;           // [NN*FIN]   = 1,000,000
static const size_t OFF_DEG  = 1000000;     // [NN]       =    50,000
static const size_t OFF_H1   = 1050000;     // [NN*HID]   = 6,400,000
static const size_t OFF_AGG2 = 7450000;     // [NN*HID]   (also holds h2)
static const size_t OFF_GSUM = 13850000;    // [NG*HID]   =    12,800
static const size_t OFF_GCNT = 13862800;    // [NG] (pad to 112)
static const size_t OFF_W1LT = 13862912;    // [FIN*HID]  = 2,560
static const size_t OFF_W1RT = 13865472;    // [FIN*HID]  = 2,560
static const size_t OFF_W2LT = 13868032;    // [HID*HID]  = 16,384
static const size_t OFF_W2RT = 13884416;    // [HID*HID]  = 16,384
// total = 13,900,800 floats = 55.6 MB

// ---------------- weight transpose: W[rows,cols] -> WT[cols,rows] ----------
__global__ void sage_transpose(const float* __restrict__ W, float* __restrict__ WT,
                               int rows, int cols) {
  int t = blockIdx.x * blockDim.x + threadIdx.x;
  if (t >= rows * cols) return;
  int r = t / cols, c = t % cols;
  WT[c * rows + r] = W[r * cols + c];
}

// ---------------- edge scatter, layer 1: agg1[dst] += x[src], deg[dst]++ ---
__global__ void sage_scatter1(const int* __restrict__ src, const int* __restrict__ dst,
                              const float* __restrict__ x,
                              float* __restrict__ agg1, float* __restrict__ deg) {
  int e = blockIdx.x * blockDim.x + threadIdx.x;
  if (e >= NE) return;
  int s = src[e], d = dst[e];
  atomicAdd(&deg[d], 1.0f);
  const float* xs = x + s * FIN;
  float* ad = agg1 + d * FIN;
  #pragma unroll
  for (int f = 0; f < FIN; ++f) atomicAdd(&ad[f], xs[f]);
}

// ---------------- layer 1: h1 = relu(mean1 @ W1lT + b1l + x @ W1rT) --------
// block = 256 thr = 8 waves; block owns 16 nodes; wave w owns out cols [16w,16w+16)
__global__ __launch_bounds__(256) void sage_layer1_wmma(
    const float* __restrict__ x, const float* __restrict__ agg1,
    const float* __restrict__ deg, const float* __restrict__ W1lT,
    const float* __restrict__ W1rT, const float* __restrict__ b1l,
    float* __restrict__ h1) {
  const int lane = threadIdx.x & 31;
  const int wave = threadIdx.x >> 5;
  const int m    = lane & 15;
  const int hi   = lane >> 4;       // 0 or 1
  const int koff = hi * 2;          // K sub-offset per A/B f32 layout
  const int nodeBase = blockIdx.x * 16;
  const int row  = nodeBase + m;
  const int ocol = wave * 16 + m;

  const float rdeg = 1.0f / fmaxf(deg[row], 1.0f);
  const float* aRow = agg1 + row * FIN;
  const float* xRow = x + row * FIN;

  v8f c = {0.f, 0.f, 0.f, 0.f, 0.f, 0.f, 0.f, 0.f};

  #pragma unroll
  for (int k0 = 0; k0 < FIN; k0 += 4) {      // mean @ W1lT
    const int k = k0 + koff;                 // even -> 8B aligned v2f load
    v2f a = *(const v2f*)(aRow + k);
    a.x *= rdeg; a.y *= rdeg;
    v2f b; b.x = W1lT[k * HID + ocol]; b.y = W1lT[(k + 1) * HID + ocol];
    c = __builtin_amdgcn_wmma_f32_16x16x4_f32(false, a, false, b, (short)0, c, false, false);
  }
  #pragma unroll
  for (int k0 = 0; k0 < FIN; k0 += 4) {      // x @ W1rT (accumulate)
    const int k = k0 + koff;
    v2f a = *(const v2f*)(xRow + k);
    v2f b; b.x = W1rT[k * HID + ocol]; b.y = W1rT[(k + 1) * HID + ocol];
    c = __builtin_amdgcn_wmma_f32_16x16x4_f32(false, a, false, b, (short)0, c, false, false);
  }

  const float bias = b1l[ocol];
  const int rbase = nodeBase + hi * 8;       // C/D layout: vgpr r -> row r (+8 for hi lanes)
  #pragma unroll
  for (int r = 0; r < 8; ++r)
    h1[(rbase + r) * HID + ocol] = fmaxf(c[r] + bias, 0.0f);
}

// ---------------- edge scatter, layer 2: agg2[dst] += h1[src] --------------
__global__ void sage_scatter2(const int* __restrict__ src, const int* __restrict__ dst,
                              const float* __restrict__ h1, float* __restrict__ agg2) {
  int t = blockIdx.x * blockDim.x + threadIdx.x;     // NE * 32 threads
  int e = t >> 5;
  if (e >= NE) return;
  int c4 = (t & 31) * 4;
  int s = src[e], d = dst[e];
  const float4 v = *(const float4*)(h1 + s * HID + c4);
  float* ad = agg2 + d * HID + c4;
  atomicAdd(&ad[0], v.x);
  atomicAdd(&ad[1], v.y);
  atomicAdd(&ad[2], v.z);
  atomicAdd(&ad[3], v.w);
}

// ---------------- layer 2: h2 = relu(mean2 @ W2lT + b2l + h1 @ W2rT) -------
// h2 written in-place into agg2 (block only touches its own 16 rows;
// __syncthreads separates all reads from stores)
__global__ __launch_bounds__(256) void sage_layer2_wmma(
    const float* __restrict__ h1, float* __restrict__ agg2,
    const float* __restrict__ deg, const float* __restrict__ W2lT,
    const float* __restrict__ W2rT, const float* __restrict__ b2l) {
  const int lane = threadIdx.x & 31;
  const int wave = threadIdx.x >> 5;
  const int m    = lane & 15;
  const int hi   = lane >> 4;
  const int koff = hi * 2;
  const int nodeBase = blockIdx.x * 16;
  const int row  = nodeBase + m;
  const int ocol = wave * 16 + m;

  const float rdeg = 1.0f / fmaxf(deg[row], 1.0f);
  const float* aRow = agg2 + row * HID;
  const float* xRow = h1 + row * HID;

  v8f c = {0.f, 0.f, 0.f, 0.f, 0.f, 0.f, 0.f, 0.f};

  #pragma unroll 8
  for (int k0 = 0; k0 < HID; k0 += 4) {      // mean2 @ W2lT
    const int k = k0 + koff;
    v2f a = *(const v2f*)(aRow + k);
    a.x *= rdeg; a.y *= rdeg;
    v2f b; b.x = W2lT[k * HID + ocol]; b.y = W2lT[(k + 1) * HID + ocol];
    c = __builtin_amdgcn_wmma_f32_16x16x4_f32(false, a, false, b, (short)0, c, false, false);
  }
  #pragma unroll 8
  for (int k0 = 0; k0 < HID; k0 += 4) {      // h1 @ W2rT (accumulate)
    const int k = k0 + koff;
    v2f a = *(const v2f*)(xRow + k);
    v2f b; b.x = W2rT[k * HID + ocol]; b.y = W2rT[(k + 1) * HID + ocol];
    c = __builtin_amdgcn_wmma_f32_16x16x4_f32(false, a, false, b, (short)0, c, false, false);
  }

  __syncthreads();                            // all agg2 reads done before overwrite

  const float bias = b2l[ocol];
  const int rbase = nodeBase + hi * 8;
  #pragma unroll
  for (int r = 0; r < 8; ++r)
    agg2[(rbase + r) * HID + ocol] = fmaxf(c[r] + bias, 0.0f);  // h2
}

// ---------------- graph mean pool: gsum[batch[i]] += h2[i], gcnt++ ---------
__global__ void sage_pool(const int* __restrict__ batch, const float* __restrict__ h2,
                          float* __restrict__ gsum, float* __restrict__ gcnt) {
  int t = blockIdx.x * blockDim.x + threadIdx.x;    // NN * 32 threads
  int i = t >> 5;
  if (i >= NN) return;
  int c4 = (t & 31) * 4;
  int g = batch[i];
  const float4 v = *(const float4*)(h2 + i * HID + c4);
  float* gd = gsum + g * HID + c4;
  atomicAdd(&gd[0], v.x);
  atomicAdd(&gd[1], v.y);
  atomicAdd(&gd[2], v.z);
  atomicAdd(&gd[3], v.w);
  if (c4 == 0) atomicAdd(&gcnt[g], 1.0f);
}

// ---------------- classifier: out = (gsum/cnt) @ Wlin^T + blin -------------
__global__ void sage_final(const float* __restrict__ gsum, const float* __restrict__ gcnt,
                           const float* __restrict__ Wlin, const float* __restrict__ blin,
                           float* __restrict__ out) {
  int t = blockIdx.x * blockDim.x + threadIdx.x;
  if (t >= NG * NOUT) return;
  int g = t / NOUT, o = t % NOUT;
  float cnt = fmaxf(gcnt[g], 1.0f);
  const float* gr = gsum + g * HID;
  const float* wr = Wlin + o * HID;
  float acc = 0.0f;
  #pragma unroll 8
  for (int k = 0; k < HID; ++k) acc += gr[k] * wr[k];
  out[t] = acc / cnt + blin[o];
}

// ---------------------------------------------------------------------------
extern "C" void kernel_launch(void* const* d_in, const int* in_sizes, int n_in,
                              void* d_out, int out_size, void* d_ws, size_t ws_size,
                              hipStream_t stream) {
  const float* x    = (const float*)d_in[0];
  const int*   ei   = (const int*)d_in[1];       // [2,E] flat: src then dst
  const int*   bat  = (const int*)d_in[2];
  const float* W1l  = (const float*)d_in[3];
  const float* b1l  = (const float*)d_in[4];
  const float* W1r  = (const float*)d_in[5];
  const float* W2l  = (const float*)d_in[6];
  const float* b2l  = (const float*)d_in[7];
  const float* W2r  = (const float*)d_in[8];
  const float* Wlin = (const float*)d_in[9];
  const float* blin = (const float*)d_in[10];
  float* out = (float*)d_out;

  float* ws   = (float*)d_ws;
  float* agg1 = ws + OFF_AGG1;
  float* deg  = ws + OFF_DEG;
  float* h1   = ws + OFF_H1;
  float* agg2 = ws + OFF_AGG2;   // becomes h2 after layer2
  float* gsum = ws + OFF_GSUM;
  float* gcnt = ws + OFF_GCNT;
  float* W1lT = ws + OFF_W1LT;
  float* W1rT = ws + OFF_W1RT;
  float* W2lT = ws + OFF_W2LT;
  float* W2rT = ws + OFF_W2RT;

  const int* src = ei;
  const int* dst = ei + NE;

  // zero accumulators every call (graph-capture-safe)
  hipMemsetAsync(agg1, 0, (size_t)(NN * FIN + NN) * sizeof(float), stream);       // agg1+deg
  hipMemsetAsync(agg2, 0, (size_t)(NN * HID) * sizeof(float), stream);
  hipMemsetAsync(gsum, 0, (size_t)(NG * HID + 112) * sizeof(float), stream);      // gsum+gcnt

  // k-major weight transposes (coalesced B operands for WMMA)
  sage_transpose<<<(HID * FIN + 255) / 256, 256, 0, stream>>>(W1l, W1lT, HID, FIN);
  sage_transpose<<<(HID * FIN + 255) / 256, 256, 0, stream>>>(W1r, W1rT, HID, FIN);
  sage_transpose<<<(HID * HID + 255) / 256, 256, 0, stream>>>(W2l, W2lT, HID, HID);
  sage_transpose<<<(HID * HID + 255) / 256, 256, 0, stream>>>(W2r, W2rT, HID, HID);

  sage_scatter1<<<(NE + 255) / 256, 256, 0, stream>>>(src, dst, x, agg1, deg);

  sage_layer1_wmma<<<NN / 16, 256, 0, stream>>>(x, agg1, deg, W1lT, W1rT, b1l, h1);

  sage_scatter2<<<(NE * 32) / 256, 256, 0, stream>>>(src, dst, h1, agg2);

  sage_layer2_wmma<<<NN / 16, 256, 0, stream>>>(h1, agg2, deg, W2lT, W2rT, b2l);

  sage_pool<<<(NN * 32 + 255) / 256, 256, 0, stream>>>(bat, agg2, gsum, gcnt);

  sage_final<<<1, 256, 0, stream>>>(gsum, gcnt, Wlin, blin, out);
}